// AdaptiveGraphGenerator_28355374088768
// MI455X (gfx1250) — compile-verified
//
#include <hip/hip_runtime.h>
#include <math.h>

// ---------------------------------------------------------------------------
// AdaptiveGraphGenerator for MI455X (gfx1250, wave32, WMMA).
//
// Live compute: node encoder (two small GEMMs + LayerNorm + exact GELU).
// The pair/edge predictor in the reference is dead code: gumbel_softmax over a
// singleton axis == 1.0, so adj = (1.0 > threshold) broadcast -> constant fill.
//
// GEMMs use V_WMMA_F32_16X16X4_F32 (exact f32 math, matches the f32 reference).
// ---------------------------------------------------------------------------

typedef __attribute__((ext_vector_type(2))) float v2f;
typedef __attribute__((ext_vector_type(8))) float v8f;

#define DIM    128
#define HID    256
#define NNODES 1024
#define HPAD   (HID + 4)   // LDS row pad: row stride 260 -> bank step 4 -> no conflicts

__device__ __forceinline__ float gelu_erf(float x) {
    // exact (erf) GELU, matching torch / jax approximate=False
    return 0.5f * x * (1.0f + erff(x * 0.70710678118654752440f));
}

// ---------------------------------------------------------------------------
// adj[1,1024,1024,1] = (1.0 > threshold) ? 1 : 0   (constant fill, 4 MB)
// ---------------------------------------------------------------------------
__global__ void __launch_bounds__(256)
adj_fill_kernel(float* __restrict__ adj, const float* __restrict__ threshold) {
    const float val = (1.0f > threshold[0]) ? 1.0f : 0.0f;
    const int i = blockIdx.x * blockDim.x + threadIdx.x;   // one float4 per thread
    float4 v;
    v.x = val; v.y = val; v.z = val; v.w = val;
    reinterpret_cast<float4*>(adj)[i] = v;
}

// ---------------------------------------------------------------------------
// node_feats = gelu(LN(x @ W1 + b1)) @ W2 + b2
// One block handles 16 rows; 256 threads = 8 wave32 waves.
// ---------------------------------------------------------------------------
__global__ void __launch_bounds__(256)
node_encoder_kernel(const float* __restrict__ x,
                    const float* __restrict__ W1, const float* __restrict__ b1,
                    const float* __restrict__ ln_g, const float* __restrict__ ln_b,
                    const float* __restrict__ W2, const float* __restrict__ b2,
                    float* __restrict__ node_feats)
{
    __shared__ float h_s[16][HPAD];     // hidden activations (16 x 256, padded)
    __shared__ float red_s[16][16];     // partial sums
    __shared__ float red2_s[16][16];    // partial sums of squares
    __shared__ float mean_s[16];
    __shared__ float rstd_s[16];

    const int tid  = threadIdx.x;
    const int wave = tid >> 5;          // 0..7
    const int lane = tid & 31;

    // WMMA f32 16x16x4 per-lane fragment geometry (ISA 7.12.2, wave32):
    //   A: lane holds row (lane&15), K-pair starting at 2*(lane>>4)
    //   B: lane holds col (lane&15), K-pair starting at 2*(lane>>4)
    //   C/D: 8 VGPRs, element r -> M = r + 8*(lane>>4), N = lane&15
    const int frow    = lane & 15;
    const int ko      = (lane >> 4) * 2;   // 0 or 2
    const int rowbase = (lane >> 4) * 8;   // 0 or 8

    const int mblk = blockIdx.x * 16;      // first node row of this block

    // ---- GEMM1: h[16,256] = x[16,128] @ W1[128,256] + b1 ------------------
    // Each wave computes two 16-wide N tiles (8 waves x 2 = 16 tiles = 256).
    {
        const float* xrow = x + (size_t)(mblk + frow) * DIM;
        #pragma unroll
        for (int t = 0; t < 2; ++t) {
            const int ntile = wave * 2 + t;        // 0..15
            const int ncol  = ntile * 16 + frow;   // this lane's B/D column
            v8f acc = {};
            #pragma unroll 4
            for (int kk = 0; kk < DIM / 4; ++kk) {  // 32 chained WMMAs
                const int k = kk * 4 + ko;
                v2f a, bfrag;
                a.x     = xrow[k];
                a.y     = xrow[k + 1];
                bfrag.x = W1[(size_t)(k    ) * HID + ncol];
                bfrag.y = W1[(size_t)(k + 1) * HID + ncol];
                acc = __builtin_amdgcn_wmma_f32_16x16x4_f32(
                        false, a, false, bfrag, (short)0, acc, false, false);
            }
            const float bias = b1[ncol];
            #pragma unroll
            for (int r = 0; r < 8; ++r)
                h_s[rowbase + r][ncol] = acc[r] + bias;
        }
    }
    __syncthreads();

    // ---- LayerNorm(256) + exact GELU, in place in LDS ---------------------
    {
        const int r = tid >> 4;   // row 0..15
        const int s = tid & 15;   // 16-element segment
        float sum = 0.0f, sumsq = 0.0f;
        #pragma unroll
        for (int j = 0; j < 16; ++j) {
            const float v = h_s[r][s * 16 + j];
            sum   += v;
            sumsq += v * v;
        }
        red_s[r][s]  = sum;
        red2_s[r][s] = sumsq;
    }
    __syncthreads();
    if (tid < 16) {
        float sum = 0.0f, sumsq = 0.0f;
        #pragma unroll
        for (int j = 0; j < 16; ++j) { sum += red_s[tid][j]; sumsq += red2_s[tid][j]; }
        const float m   = sum * (1.0f / HID);
        const float var = sumsq * (1.0f / HID) - m * m;
        mean_s[tid] = m;
        rstd_s[tid] = rsqrtf(var + 1e-5f);
    }
    __syncthreads();
    {
        const int r = tid >> 4;
        const int s = tid & 15;
        const float m  = mean_s[r];
        const float rs = rstd_s[r];
        #pragma unroll
        for (int j = 0; j < 16; ++j) {
            const int c = s * 16 + j;
            const float v = (h_s[r][c] - m) * rs * ln_g[c] + ln_b[c];
            h_s[r][c] = gelu_erf(v);
        }
    }
    __syncthreads();

    // ---- GEMM2: node_feats[16,128] = gelu_h[16,256] @ W2[256,128] + b2 ----
    // One 16-wide N tile per wave (8 waves x 16 = 128).
    {
        const int ntile = wave;
        const int ncol  = ntile * 16 + frow;
        v8f acc = {};
        #pragma unroll 4
        for (int kk = 0; kk < HID / 4; ++kk) {  // 64 chained WMMAs
            const int k = kk * 4 + ko;
            v2f a, bfrag;
            a.x     = h_s[frow][k];
            a.y     = h_s[frow][k + 1];
            bfrag.x = W2[(size_t)(k    ) * DIM + ncol];
            bfrag.y = W2[(size_t)(k + 1) * DIM + ncol];
            acc = __builtin_amdgcn_wmma_f32_16x16x4_f32(
                    false, a, false, bfrag, (short)0, acc, false, false);
        }
        const float bias = b2[ncol];
        #pragma unroll
        for (int r = 0; r < 8; ++r)
            node_feats[(size_t)(mblk + rowbase + r) * DIM + ncol] = acc[r] + bias;
    }
}

// ---------------------------------------------------------------------------
extern "C" void kernel_launch(void* const* d_in, const int* in_sizes, int n_in,
                              void* d_out, int out_size, void* d_ws, size_t ws_size,
                              hipStream_t stream) {
    const float* x    = (const float*)d_in[0];
    const float* W1   = (const float*)d_in[1];
    const float* b1   = (const float*)d_in[2];
    const float* ln_g = (const float*)d_in[3];
    const float* ln_b = (const float*)d_in[4];
    const float* W2   = (const float*)d_in[5];
    const float* b2   = (const float*)d_in[6];
    // d_in[7..10] (edge predictor weights) are dead code in the reference.
    const float* thr  = (const float*)d_in[11];

    float* out        = (float*)d_out;
    float* adj        = out;                                  // [1,1024,1024,1]
    float* node_feats = out + (size_t)NNODES * NNODES;        // [1,1024,128]

    // 4 MB constant fill: 1048576 floats / 4 per thread / 256 per block
    adj_fill_kernel<<<(NNODES * NNODES / 4) / 256, 256, 0, stream>>>(adj, thr);

    // 64 blocks x 16 rows
    node_encoder_kernel<<<NNODES / 16, 256, 0, stream>>>(
        x, W1, b1, ln_g, ln_b, W2, b2, node_feats);
}